// Loss_21981642621156
// MI455X (gfx1250) — compile-verified
//
#include <hip/hip_runtime.h>
#include <math.h>

// ---------------------------------------------------------------------------
// Loss kernel: gather 3 probs per (l,b), p += eps if p<eps, -log(p)*mask,
// sum over all 4096 items, divide by B=32.  Single workgroup (32 waves on one
// WGP) -> deterministic reduction, enough waves to hide random-gather latency.
// Cross-lane reduction uses V_WMMA_F32_16X16X4_F32 (A = lane partials,
// B = ones) so the whole accumulation stays in f32 on the matrix pipe.
// ---------------------------------------------------------------------------

typedef __attribute__((ext_vector_type(2))) float v2f;
typedef __attribute__((ext_vector_type(8))) float v8f;

#define L_A        128
#define NB         32
#define N_ITEMS    (L_A * NB)      // 4096
#define NUM_RULES  1024
#define NUM_TOKENS 32000
#define REF_LEN    512
#define BLOCK      1024

// Reduce one f32 value per lane across the full wave32 using
// v_wmma_f32_16x16x4_f32:  D = A x ones + 0.
// A layout (32-bit A 16x4): lanes 0-15: VGPR0->K=0, VGPR1->K=1;
//                           lanes 16-31: VGPR0->K=2, VGPR1->K=3.
// With A.vgpr0 = s_lane, A.vgpr1 = 0 and B = all ones:
//   D[m,n] = s_m + s_{m+16}  for every n.
// D layout: VGPR j holds D[M=j, N=lane] (lanes 0-15), D[M=j+8, .] (lanes 16-31),
// so summing the 8 D VGPRs gives  sum(r_0..r_7)  on lanes 0-15 and
// sum(r_8..r_15) on lanes 16-31; one xor-16 shuffle completes the total.
__device__ __forceinline__ float wave_reduce_f32_wmma(float s) {
  v2f a; a[0] = s;    a[1] = 0.0f;   // partials in K=0 / K=2 slots only
  v2f b; b[0] = 1.0f; b[1] = 1.0f;   // B = ones (layout-independent)
  v8f c = {};
  c = __builtin_amdgcn_wmma_f32_16x16x4_f32(
      /*neg_a=*/false, a, /*neg_b=*/false, b,
      /*c_mod=*/(short)0, c, /*reuse_a=*/false, /*reuse_b=*/false);
  float t = ((c[0] + c[1]) + (c[2] + c[3])) + ((c[4] + c[5]) + (c[6] + c[7]));
  t += __shfl_xor(t, 16, 32);        // combine M=0..7 half with M=8..15 half
  return t;                          // full wave sum, broadcast to all lanes
}

__global__ __launch_bounds__(BLOCK)
void Loss_21981642621156_kernel(const float* __restrict__ rule_probs,
                                const float* __restrict__ token_probs,
                                const float* __restrict__ reference_probs,
                                const int*   __restrict__ gt_actions,
                                const int*   __restrict__ gt_mask,
                                float*       __restrict__ out) {
  __shared__ float wave_sum[BLOCK / 32];

  const int tid = threadIdx.x;
  float s = 0.0f;

#pragma unroll
  for (int it = 0; it < N_ITEMS / BLOCK; ++it) {
    const int id = tid + it * BLOCK;          // id = l*32 + b, row-major (L_A,B)

    const int ir = gt_actions[id * 3 + 0];
    const int it2 = gt_actions[id * 3 + 1];
    const int ig = gt_actions[id * 3 + 2];

    // branchless safe gathers: clamp index to 0, zero the lane if idx < 0
    const int sr = ir  < 0 ? 0 : ir;
    const int st = it2 < 0 ? 0 : it2;
    const int sg = ig  < 0 ? 0 : ig;

    float pr = rule_probs     [(size_t)id * NUM_RULES  + sr];
    float pt = token_probs    [(size_t)id * NUM_TOKENS + st];
    float pg = reference_probs[(size_t)id * REF_LEN    + sg];
    if (ir  < 0) pr = 0.0f;
    if (it2 < 0) pt = 0.0f;
    if (ig  < 0) pg = 0.0f;

    float p = pr + pt + pg;
    const float eps = 1e-7f;
    if (p < eps) p += eps;                    // reference ADDS eps, not clamps

    const float m = (gt_mask[id] != 0) ? 1.0f : 0.0f;
    s += -logf(p) * m;
  }

  // stage 1: wave32 reduction on the matrix pipe
  const float wtot = wave_reduce_f32_wmma(s);
  const int lane = tid & 31;
  const int wave = tid >> 5;
  if (lane == 0) wave_sum[wave] = wtot;
  __syncthreads();

  // stage 2: wave 0 reduces the 32 wave partials (EXEC all-ones inside branch)
  if (tid < 32) {
    const float tot = wave_reduce_f32_wmma(wave_sum[tid]);
    if (tid == 0) out[0] = tot * (1.0f / (float)NB);   // mean over B
  }
}

extern "C" void kernel_launch(void* const* d_in, const int* in_sizes, int n_in,
                              void* d_out, int out_size, void* d_ws, size_t ws_size,
                              hipStream_t stream) {
  (void)in_sizes; (void)n_in; (void)out_size; (void)d_ws; (void)ws_size;
  const float* rule_probs      = (const float*)d_in[0];
  const float* token_probs     = (const float*)d_in[1];
  const float* reference_probs = (const float*)d_in[2];
  const int*   gt_actions      = (const int*)d_in[3];
  const int*   gt_mask         = (const int*)d_in[4];
  float*       out             = (float*)d_out;

  Loss_21981642621156_kernel<<<1, BLOCK, 0, stream>>>(
      rule_probs, token_probs, reference_probs, gt_actions, gt_mask, out);
}